// Sampler_39118562132262
// MI455X (gfx1250) — compile-verified
//
#include <hip/hip_runtime.h>
#include <hip/hip_bf16.h>
#include <math.h>

#ifndef __has_builtin
#define __has_builtin(x) 0
#endif

static constexpr int   B_ROWS   = 256;
static constexpr int   VOCAB    = 128000;
static constexpr int   NT       = 1024;             // 32 wave32s
static constexpr int   VPT      = 4;                // floats per thread per chunk (b128)
static constexpr int   CHUNK    = NT * VPT;         // 4096 floats = 16KB
static constexpr int   NCHUNK   = (VOCAB + CHUNK - 1) / CHUNK; // 32 (tail partial)
static constexpr int   NLOGPROB = 20;
static constexpr int   LISTCAP  = 192;
static constexpr int   NREP     = 4;                // histogram replicas (contention relief)
static constexpr float SAMPLING_EPS = 1e-5f;
static constexpr float RMAX     = 0.9999999403953552f;
static constexpr float RMAX_LOG = -5.960464477539063e-08f;

// ---- order-preserving float <-> uint key -------------------------------
static __device__ __forceinline__ unsigned key_of(float x) {
  unsigned u = __float_as_uint(x);
  return (u & 0x80000000u) ? ~u : (u | 0x80000000u);
}
static __device__ __forceinline__ float val_of(unsigned k) {
  unsigned u = (k & 0x80000000u) ? (k & 0x7fffffffu) : ~k;
  return __uint_as_float(u);
}

// ---- CDNA5 async global -> LDS copy (16B per lane) ---------------------
typedef int b128_t __attribute__((vector_size(16)));

static __device__ __forceinline__ void async_copy_b128(const float* g, float* l) {
#if defined(__gfx1250__) && __has_builtin(__builtin_amdgcn_global_load_async_to_lds_b128)
  __builtin_amdgcn_global_load_async_to_lds_b128((b128_t*)g, (b128_t*)l, 0, 0);
#else
  float4 v = *reinterpret_cast<const float4*>(g);
  *reinterpret_cast<float4*>(l) = v;
#endif
}
static __device__ __forceinline__ void wait_async(int n) {
#if defined(__gfx1250__)
#if __has_builtin(__builtin_amdgcn_s_wait_asynccnt)
  if (n == 0) __builtin_amdgcn_s_wait_asynccnt(0);
  else        __builtin_amdgcn_s_wait_asynccnt(1);
#else
  if (n == 0) asm volatile("s_wait_asynccnt 0" ::: "memory");
  else        asm volatile("s_wait_asynccnt 1" ::: "memory");
#endif
#endif
  (void)n;
}

// ---- stream one row through LDS with async double-buffering ------------
// Each thread reads back only the 16B it loaded -> no block barrier needed,
// only the per-wave ASYNCcnt wait.
template <typename F>
static __device__ __forceinline__ void stream_row(const float* __restrict__ rowp,
                                                  float* __restrict__ stage, F&& f) {
  const int tid = threadIdx.x;
  {
    const int base0 = tid * VPT;
    if (base0 < VOCAB) async_copy_b128(rowp + base0, stage + tid * VPT);
  }
  for (int c = 0; c < NCHUNK; ++c) {
    if (c + 1 < NCHUNK) {
      const int nbase = (c + 1) * CHUNK + tid * VPT;
      if (nbase < VOCAB)
        async_copy_b128(rowp + nbase, stage + ((c + 1) & 1) * CHUNK + tid * VPT);
      wait_async(1);
    } else {
      wait_async(0);
    }
    const int base = c * CHUNK + tid * VPT;
    if (base < VOCAB) {
      const float* lp = stage + (c & 1) * CHUNK + tid * VPT;
      f(base, lp[0], lp[1], lp[2], lp[3]);
    }
  }
}

// ---- two-level crossing scans (64 parallel group sums + <=128 serial) --
// Descending (counts): find d s.t. sum of buckets above d < k <= sum incl d.
static __device__ __forceinline__ void scan_desc(const unsigned* h, int nb, unsigned k,
                                                 unsigned* sup, unsigned* s_u, int tid) {
  const int G = 64, gs = nb / G;
  if (tid < G) {
    unsigned s = 0;
    for (int i = 0; i < gs; ++i) s += h[tid * gs + i];
    sup[tid] = s;
  }
  __syncthreads();
  if (tid == 0) {
    unsigned cum = 0; int g = G - 1;
    for (; g > 0; --g) { unsigned s = sup[g]; if (cum + s >= k) break; cum += s; }
    int d = g * gs + gs - 1;
    for (; d > g * gs; --d) { unsigned hh = h[d]; if (cum + hh >= k) break; cum += hh; }
    s_u[0] = (unsigned)d; s_u[1] = k - cum;
  }
  __syncthreads();
}
// Ascending (weighted): find d s.t. mass below d <= target < mass incl d.
static __device__ __forceinline__ void scan_asc_f(const float* h, int nb, float target,
                                                  float* supf, unsigned* s_u, float* s_f, int tid) {
  const int G = 64, gs = nb / G;
  if (tid < G) {
    float s = 0.0f;
    for (int i = 0; i < gs; ++i) s += h[tid * gs + i];
    supf[tid] = s;
  }
  __syncthreads();
  if (tid == 0) {
    float cum = 0.0f; int g = 0;
    for (; g < G - 1; ++g) { float s = supf[g]; if (cum + s > target) break; cum += s; }
    int d = g * gs;
    for (; d < g * gs + gs - 1; ++d) { float hh = h[d]; if (cum + hh > target) break; cum += hh; }
    s_u[0] = (unsigned)d; s_f[0] = target - cum;
  }
  __syncthreads();
}

__global__ __launch_bounds__(NT)
void Sampler_39118562132262_kernel(const float* __restrict__ logits,
                                   const float* __restrict__ temperature,
                                   const int*   __restrict__ top_k,
                                   const float* __restrict__ top_p,
                                   const float* __restrict__ gumbel_u,
                                   float* __restrict__ out, int nrows) {
  __shared__ float              s_stage[2 * CHUNK];      // 32 KB
  __shared__ unsigned           s_histR[NREP * 4096];    // 64 KB (uint/float aliased)
  __shared__ unsigned           s_sup[64];
  __shared__ float              s_supf[64];
  __shared__ unsigned long long s_list[LISTCAP];
  __shared__ unsigned long long s_max64;
  __shared__ unsigned           s_u[4];
  __shared__ float              s_f[4];
  __shared__ unsigned           s_cnt, s_cnt2, s_maxkey;
  __shared__ float              s_acc;

  const int row = blockIdx.x;
  if (row >= nrows) return;
  const int tid = threadIdx.x;
  const unsigned rep = ((unsigned)tid >> 5) & (NREP - 1);  // wave-id mod NREP
  unsigned* h  = s_histR;                                   // replica 0 (scan target)
  unsigned* hr = s_histR + rep * 4096;                      // this wave's replica
  float*    fh  = reinterpret_cast<float*>(s_histR);
  float*    fhr = reinterpret_cast<float*>(s_histR) + rep * 4096;

  const float* rowp  = logits   + (size_t)row * VOCAB;
  const float* gurow = gumbel_u + (size_t)row * VOCAB;

  const float traw   = temperature[row];
  const bool  greedy = traw < SAMPLING_EPS;
  const float invt   = greedy ? 1.0f : (1.0f / traw);
  int kk = top_k[row];
  kk = kk < 1 ? 1 : (kk > VOCAB ? VOCAB : kk);
  const float pp = top_p[row];

  // ============ P1: row max + top-k radix level 1 (bits 31:20) ==========
  for (int i = tid; i < NREP * 4096; i += NT) s_histR[i] = 0u;
  if (tid == 0) s_maxkey = 0u;
  __syncthreads();
  unsigned lmax = 0u;
  stream_row(rowp, s_stage, [&](int base, float a, float b, float c, float d) {
    (void)base;
    float x[4] = {a * invt, b * invt, c * invt, d * invt};
#pragma unroll
    for (int j = 0; j < 4; ++j) {
      unsigned k = key_of(x[j]);
      atomicAdd(&hr[k >> 20], 1u);
      lmax = lmax > k ? lmax : k;
    }
  });
  atomicMax(&s_maxkey, lmax);
  __syncthreads();
  for (int i = tid; i < 4096; i += NT)                     // reduce replicas -> h
    h[i] = h[i] + h[i + 4096] + h[i + 2 * 4096] + h[i + 3 * 4096];
  __syncthreads();
  scan_desc(h, 4096, (unsigned)kk, s_sup, s_u, tid);
  const unsigned d1 = s_u[0];
  unsigned krem = s_u[1];
  const float m = val_of(s_maxkey);

  // ============ P2: top-k radix level 2 (bits 19:8) =====================
  for (int i = tid; i < 4096; i += NT) h[i] = 0u;
  __syncthreads();
  stream_row(rowp, s_stage, [&](int base, float a, float b, float c, float d) {
    (void)base;
    float x[4] = {a * invt, b * invt, c * invt, d * invt};
#pragma unroll
    for (int j = 0; j < 4; ++j) {
      unsigned k = key_of(x[j]);
      if ((k >> 20) == d1) atomicAdd(&h[(k >> 8) & 0xFFFu], 1u);
    }
  });
  __syncthreads();
  scan_desc(h, 4096, krem, s_sup, s_u, tid);
  const unsigned d2 = s_u[0];
  krem = s_u[1];
  const unsigned pre24 = (d1 << 12) | d2;

  // ============ P3: top-k radix level 3 (bits 7:0) ======================
  for (int i = tid; i < 256; i += NT) h[i] = 0u;
  __syncthreads();
  stream_row(rowp, s_stage, [&](int base, float a, float b, float c, float d) {
    (void)base;
    float x[4] = {a * invt, b * invt, c * invt, d * invt};
#pragma unroll
    for (int j = 0; j < 4; ++j) {
      unsigned k = key_of(x[j]);
      if ((k >> 8) == pre24) atomicAdd(&h[k & 0xFFu], 1u);
    }
  });
  __syncthreads();
  scan_desc(h, 256, krem, s_sup, s_u, tid);
  const unsigned tk_u = (pre24 << 8) | s_u[0];   // exact k-th largest key

  // ============ P4: S1 + weighted (exp) radix level 1 for top-p =========
  for (int i = tid; i < NREP * 4096; i += NT) fh[i] = 0.0f;
  if (tid == 0) s_acc = 0.0f;
  __syncthreads();
  float lsum = 0.0f;
  stream_row(rowp, s_stage, [&](int base, float a, float b, float c, float d) {
    (void)base;
    float x[4] = {a * invt, b * invt, c * invt, d * invt};
#pragma unroll
    for (int j = 0; j < 4; ++j) {
      unsigned k = key_of(x[j]);
      if (k >= tk_u) { float w = __expf(x[j] - m); lsum += w; atomicAdd(&fhr[k >> 20], w); }
    }
  });
  atomicAdd(&s_acc, lsum);
  __syncthreads();
  for (int i = tid; i < 4096; i += NT)
    fh[i] = fh[i] + fh[i + 4096] + fh[i + 2 * 4096] + fh[i + 3 * 4096];
  __syncthreads();
  const float S1 = s_acc;
  scan_asc_f(fh, 4096, (1.0f - pp) * S1, s_supf, s_u, s_f, tid);
  const unsigned wd1 = s_u[0];
  float wtarget = s_f[0];

  // ============ P5: weighted radix level 2 ==============================
  for (int i = tid; i < 4096; i += NT) fh[i] = 0.0f;
  __syncthreads();
  stream_row(rowp, s_stage, [&](int base, float a, float b, float c, float d) {
    (void)base;
    float x[4] = {a * invt, b * invt, c * invt, d * invt};
#pragma unroll
    for (int j = 0; j < 4; ++j) {
      unsigned k = key_of(x[j]);
      if (k >= tk_u && (k >> 20) == wd1)
        atomicAdd(&fh[(k >> 8) & 0xFFFu], __expf(x[j] - m));
    }
  });
  __syncthreads();
  scan_asc_f(fh, 4096, wtarget, s_supf, s_u, s_f, tid);
  const unsigned wd2 = s_u[0];
  wtarget = s_f[0];
  const unsigned wpre24 = (wd1 << 12) | wd2;

  // ============ P6: weighted radix level 3 ==============================
  for (int i = tid; i < 256; i += NT) fh[i] = 0.0f;
  __syncthreads();
  stream_row(rowp, s_stage, [&](int base, float a, float b, float c, float d) {
    (void)base;
    float x[4] = {a * invt, b * invt, c * invt, d * invt};
#pragma unroll
    for (int j = 0; j < 4; ++j) {
      unsigned k = key_of(x[j]);
      if (k >= tk_u && (k >> 8) == wpre24)
        atomicAdd(&fh[k & 0xFFu], __expf(x[j] - m));
    }
  });
  __syncthreads();
  scan_asc_f(fh, 256, wtarget, s_supf, s_u, s_f, tid);
  const unsigned tp_u = (wpre24 << 8) | s_u[0];
  const unsigned Tu = tp_u > tk_u ? tp_u : tk_u;   // final keep threshold

  // ====== P7: S2, kept count, gumbel argmax, top-20 radix level 1 =======
  for (int i = tid; i < NREP * 4096; i += NT) s_histR[i] = 0u;
  if (tid == 0) { s_cnt = 0u; s_acc = 0.0f; s_max64 = 0ull; }
  __syncthreads();
  float lS2 = 0.0f; unsigned lc = 0u; unsigned long long lenc = 0ull;
  stream_row(rowp, s_stage, [&](int base, float a, float b, float c, float d) {
    float4 uu = *reinterpret_cast<const float4*>(gurow + base);
    float x[4] = {a * invt, b * invt, c * invt, d * invt};
    float u[4] = {uu.x, uu.y, uu.z, uu.w};
#pragma unroll
    for (int j = 0; j < 4; ++j) {
      unsigned k = key_of(x[j]);
      if (k >= Tu) {
        lS2 += __expf(x[j] - m);
        lc++;
        atomicAdd(&hr[k >> 20], 1u);
        float q  = -((u[j] >= RMAX) ? RMAX_LOG : __logf(u[j]));
        float sv = greedy ? x[j] : (x[j] - __logf(q));  // monotone in p/q
        unsigned long long enc =
            ((unsigned long long)key_of(sv) << 32) |
            (unsigned long long)(~(unsigned)(base + j));
        lenc = lenc > enc ? lenc : enc;
      }
    }
  });
  atomicAdd(&s_acc, lS2);
  atomicAdd(&s_cnt, lc);
  atomicMax(&s_max64, lenc);
  __syncthreads();
  for (int i = tid; i < 4096; i += NT)
    h[i] = h[i] + h[i + 4096] + h[i + 2 * 4096] + h[i + 3 * 4096];
  __syncthreads();
  const float    S2    = s_acc;
  const unsigned keptc = s_cnt;
  const unsigned sidx  = ~(unsigned)(s_max64 & 0xffffffffull);
  unsigned c20 = keptc < (unsigned)NLOGPROB ? keptc : (unsigned)NLOGPROB;
  if (c20 == 0u) c20 = 1u;
  scan_desc(h, 4096, c20, s_sup, s_u, tid);
  const unsigned td1 = s_u[0];
  unsigned tkrem = s_u[1];

  // ============ P8: top-20 level 2 + fetch sampled logit ================
  for (int i = tid; i < 4096; i += NT) h[i] = 0u;
  __syncthreads();
  stream_row(rowp, s_stage, [&](int base, float a, float b, float c, float d) {
    float x[4] = {a * invt, b * invt, c * invt, d * invt};
#pragma unroll
    for (int j = 0; j < 4; ++j) {
      unsigned k = key_of(x[j]);
      if ((unsigned)(base + j) == sidx) s_f[1] = x[j];  // single writer
      if (k >= Tu && (k >> 20) == td1) atomicAdd(&h[(k >> 8) & 0xFFFu], 1u);
    }
  });
  __syncthreads();
  const float    xs = s_f[1];
  const unsigned ks = key_of(xs);
  scan_desc(h, 4096, tkrem, s_sup, s_u, tid);
  const unsigned td2 = s_u[0];
  tkrem = s_u[1];
  const unsigned tpre24 = (td1 << 12) | td2;

  // ============ P9: top-20 level 3 + rank ===============================
  for (int i = tid; i < 256; i += NT) h[i] = 0u;
  if (tid == 0) s_cnt = 0u;
  __syncthreads();
  unsigned lrank = 0u;
  stream_row(rowp, s_stage, [&](int base, float a, float b, float c, float d) {
    (void)base;
    float x[4] = {a * invt, b * invt, c * invt, d * invt};
#pragma unroll
    for (int j = 0; j < 4; ++j) {
      unsigned k = key_of(x[j]);
      if (k > ks) lrank++;   // k > ks implies kept (ks >= Tu)
      if (k >= Tu && (k >> 8) == tpre24) atomicAdd(&h[k & 0xFFu], 1u);
    }
  });
  atomicAdd(&s_cnt, lrank);
  __syncthreads();
  scan_desc(h, 256, tkrem, s_sup, s_u, tid);
  const unsigned t20_u = (tpre24 << 8) | s_u[0];
  const unsigned rank  = s_cnt;

  // ============ P10: gather top-20 candidates ===========================
  if (tid == 0) s_cnt2 = 0u;
  __syncthreads();
  stream_row(rowp, s_stage, [&](int base, float a, float b, float c, float d) {
    float x[4] = {a * invt, b * invt, c * invt, d * invt};
#pragma unroll
    for (int j = 0; j < 4; ++j) {
      unsigned k = key_of(x[j]);
      if (k >= t20_u && k >= Tu) {
        unsigned pos = atomicAdd(&s_cnt2, 1u);
        if (pos < (unsigned)LISTCAP)
          s_list[pos] = ((unsigned long long)k << 32) |
                        (unsigned long long)(~(unsigned)(base + j));
      }
    }
  });
  __syncthreads();

  // ============ finalize + write outputs ================================
  if (tid == 0) {
    int n = (int)s_cnt2; if (n > LISTCAP) n = LISTCAP;
    for (int i = 0; i < n && i < NLOGPROB; ++i) {   // partial selection sort desc
      int best = i;
      for (int j = i + 1; j < n; ++j) if (s_list[j] > s_list[best]) best = j;
      unsigned long long t = s_list[i]; s_list[i] = s_list[best]; s_list[best] = t;
    }
    const float logS2   = __logf(S2);
    const float samp_lp = xs - m - logS2;
    float* out_samp = out;
    float* out_idx  = out + B_ROWS;
    float* out_lp   = out + B_ROWS + B_ROWS * (NLOGPROB + 1);
    float* out_rank = out + B_ROWS + 2 * B_ROWS * (NLOGPROB + 1);
    out_samp[row] = (float)sidx;
    out_idx[row * (NLOGPROB + 1) + 0] = (float)sidx;
    out_lp [row * (NLOGPROB + 1) + 0] = samp_lp;
    for (int i = 0; i < NLOGPROB; ++i) {
      unsigned idx; float lp;
      if (i < n) {
        unsigned long long e = s_list[i];
        unsigned k = (unsigned)(e >> 32);
        idx = ~(unsigned)(e & 0xffffffffull);
        lp  = val_of(k) - m - logS2;
      } else { idx = 0u; lp = -INFINITY; }
      out_idx[row * (NLOGPROB + 1) + 1 + i] = (float)idx;
      out_lp [row * (NLOGPROB + 1) + 1 + i] = lp;
    }
    out_rank[row] = (float)rank;
  }
}

extern "C" void kernel_launch(void* const* d_in, const int* in_sizes, int n_in,
                              void* d_out, int out_size, void* d_ws, size_t ws_size,
                              hipStream_t stream) {
  (void)n_in; (void)d_ws; (void)ws_size; (void)out_size;
  const float* logits      = (const float*)d_in[0];
  const float* temperature = (const float*)d_in[1];
  const int*   top_k       = (const int*)d_in[2];
  const float* top_p       = (const float*)d_in[3];
  const float* gumbel      = (const float*)d_in[4];
  // d_in[5] = num_logprobs (==20, compiled in as NLOGPROB)
  const int rows = in_sizes[1];  // B from temperature
  Sampler_39118562132262_kernel<<<rows, NT, 0, stream>>>(
      logits, temperature, top_k, top_p, gumbel, (float*)d_out, rows);
}